// SampledSoftmaxLoss_6588479832548
// MI455X (gfx1250) — compile-verified
//
#include <hip/hip_runtime.h>
#include <hip/hip_bf16.h>
#include <stdint.h>

typedef __attribute__((ext_vector_type(2))) float v2f;
typedef __attribute__((ext_vector_type(8))) float v8f;

#define B_ROWS   4096
#define DIM      64
#define NITEMS   1000000
#define S_COLS   40960
#define NCB      20                       // column slices for partial LSE
#define COLS_PER_CB (S_COLS / NCB)        // 2048
#define WAVES_PER_BLOCK 8
#define TILE_N   (16 * WAVES_PER_BLOCK)   // 128 columns per block iteration
#define ITERS_PER_CB (COLS_PER_CB / TILE_N) // 16

#define LDS_ROW  68                       // 64 floats + 4 pad -> conflict-free b64 reads

#define NEG_BIG  (-3.0e38f)
#define NEG_MAXF (-3.4028235e38f)
#define F32_TINY (1.17549435e-38f)

// -log(expected_count) for a log-uniform sampled id (matches reference math)
__device__ __forceinline__ float neg_log_expected(float idf, int num_tries) {
    float p = (logf(idf + 2.0f) - logf(idf + 1.0f)) / logf((float)(NITEMS) + 1.0f);
    float q = 1.0f - expf((float)num_tries * logf(1.0f - p));   // 1 - (1-p)^T
    q = fmaxf(q, F32_TINY);
    return -logf(q);
}

// CDNA5 async global->LDS copy (ASYNCcnt-tracked), 16 bytes per lane
__device__ __forceinline__ void async_copy_b128(unsigned lds_off, const void* gptr) {
    asm volatile("global_load_async_to_lds_b128 %0, %1, off"
                 :: "v"(lds_off), "v"(gptr) : "memory");
}
__device__ __forceinline__ void wait_async0() {
    asm volatile("s_wait_asynccnt 0x0" ::: "memory");
}

// -------- Kernel 1: gather sampled embeddings + per-column correction --------
__global__ void ssm_gather_kernel(const float* __restrict__ item_emb,
                                  const float* __restrict__ item_bias,
                                  const int*   __restrict__ samp_ids,
                                  const int*   __restrict__ num_tries_p,
                                  float* __restrict__ semb,
                                  float* __restrict__ corr) {
    int gid = blockIdx.x * blockDim.x + threadIdx.x;   // S*4 threads
    int s = gid >> 2;
    int q = gid & 3;
    if (s >= S_COLS) return;
    int id = samp_ids[s];
    const float4* src = (const float4*)(item_emb + (size_t)id * DIM);
    float4*       dst = (float4*)(semb + (size_t)s * DIM);
#pragma unroll
    for (int i = 0; i < 4; ++i) dst[q * 4 + i] = src[q * 4 + i];
    if (q == 0) {
        corr[s] = item_bias[id] + neg_log_expected((float)id, *num_tries_p);
    }
}

// -------- Kernel 2: positive logits (corrected) --------
__global__ void ssm_pos_kernel(const float* __restrict__ user,
                               const float* __restrict__ item_emb,
                               const float* __restrict__ item_bias,
                               const int*   __restrict__ pos_ids,
                               const int*   __restrict__ num_tries_p,
                               float* __restrict__ pos_logits) {
    int b = blockIdx.x * blockDim.x + threadIdx.x;
    if (b >= B_ROWS) return;
    int id = pos_ids[b];
    const float4* u = (const float4*)(user + (size_t)b * DIM);
    const float4* e = (const float4*)(item_emb + (size_t)id * DIM);
    float acc = 0.0f;
#pragma unroll
    for (int i = 0; i < 16; ++i) {
        float4 a = u[i], c = e[i];
        acc += a.x * c.x + a.y * c.y + a.z * c.z + a.w * c.w;
    }
    pos_logits[b] = acc + item_bias[id] + neg_log_expected((float)id, *num_tries_p);
}

// -------- Kernel 3: fused WMMA GEMM + online logsumexp partials --------
__global__ void __launch_bounds__(256)
ssm_gemm_lse_kernel(const float* __restrict__ user,
                    const float* __restrict__ semb,
                    const float* __restrict__ corr,
                    const int*   __restrict__ pos_ids,
                    const int*   __restrict__ samp_ids,
                    float* __restrict__ partM,
                    float* __restrict__ partS) {
    __shared__ float btile[2][TILE_N * LDS_ROW];   // double-buffered B tile (padded)
    __shared__ float lm[WAVES_PER_BLOCK][16];
    __shared__ float ls[WAVES_PER_BLOCK][16];

    const int rb   = blockIdx.x;          // row tile: 16 rows
    const int cb   = blockIdx.y;          // column slice: COLS_PER_CB cols
    const int wave = threadIdx.x >> 5;
    const int lane = threadIdx.x & 31;
    const int half = lane >> 4;           // selects K-pair within each 4-wide K step
    const int l15  = lane & 15;
    const int row_base = rb * 16;
    const int col_base = cb * COLS_PER_CB;

    // A matrix (16 x 64 fp32): lane l15 = row M; half selects K offset {0,1}/{2,3}.
    v2f A[16];
    {
        const float* urow = user + (size_t)(row_base + l15) * DIM + 2 * half;
#pragma unroll
        for (int k = 0; k < 16; ++k) A[k] = *(const v2f*)(urow + 4 * k);
    }

    // positive ids for the 8 rows this lane's accumulator holds (M = v + 8*half)
    int prow[8];
#pragma unroll
    for (int v = 0; v < 8; ++v) prow[v] = pos_ids[row_base + 8 * half + v];

    float m[8], s[8];
#pragma unroll
    for (int v = 0; v < 8; ++v) { m[v] = NEG_BIG; s[v] = 0.0f; }

    // Per-thread copy lambda: 8 x 16B chunks of the 128x64 B tile -> padded LDS
    auto issue_copy = [&](int buf, int colb) {
#pragma unroll
        for (int i = 0; i < 8; ++i) {
            int c    = threadIdx.x + 256 * i;    // chunk id 0..2047
            int sr   = c >> 4;                   // semb row within tile (0..127)
            int off  = c & 15;                   // 16B chunk within row
            const void* gp = (const void*)(semb + (size_t)(colb + sr) * DIM + off * 4);
            unsigned lo = (unsigned)(uintptr_t)&btile[buf][sr * LDS_ROW + off * 4];
            async_copy_b128(lo, gp);
        }
    };

    issue_copy(0, col_base);

    const float* myB = &btile[0][0] + (wave * 16 + l15) * LDS_ROW + 2 * half;

    for (int it = 0; it < ITERS_PER_CB; ++it) {
        const int buf = it & 1;
        wait_async0();          // this wave's copies for buf are done
        __syncthreads();        // => ALL waves' copies for buf are done
        if (it + 1 < ITERS_PER_CB)
            issue_copy(buf ^ 1, col_base + (it + 1) * TILE_N);

        // B fragment from LDS: conflict-free ds_load_b64 (padded rows)
        const float* bp = myB + buf * (TILE_N * LDS_ROW);
        v2f Bv[16];
#pragma unroll
        for (int k = 0; k < 16; ++k) Bv[k] = *(const v2f*)(bp + 4 * k);

        v8f c = {};
#pragma unroll
        for (int k = 0; k < 16; ++k) {
            c = __builtin_amdgcn_wmma_f32_16x16x4_f32(
                    false, A[k], false, Bv[k], (short)0, c, false, false);
        }

        const int n   = col_base + it * TILE_N + wave * 16 + l15;  // output column
        const float cn  = corr[n];
        const int   sid = samp_ids[n];
#pragma unroll
        for (int v = 0; v < 8; ++v) {
            float x = c[v] + cn;
            if (sid == prow[v]) x = NEG_MAXF;     // mask accidental hit
            float mn = fmaxf(m[v], x);
            s[v] = s[v] * expf(m[v] - mn) + expf(x - mn);
            m[v] = mn;
        }
    }

    // merge across the 16 lanes of each half (wave32 shuffles)
#pragma unroll
    for (int off = 1; off < 16; off <<= 1) {
#pragma unroll
        for (int v = 0; v < 8; ++v) {
            float mo = __shfl_xor(m[v], off, 32);
            float so = __shfl_xor(s[v], off, 32);
            float mn = fmaxf(m[v], mo);
            s[v] = s[v] * expf(m[v] - mn) + so * expf(mo - mn);
            m[v] = mn;
        }
    }

    // merge across the 8 waves of the block via LDS, then write partials
    if (l15 == 0) {   // lane 0 (rows 0..7) and lane 16 (rows 8..15)
#pragma unroll
        for (int v = 0; v < 8; ++v) {
            lm[wave][8 * half + v] = m[v];
            ls[wave][8 * half + v] = s[v];
        }
    }
    __syncthreads();
    if (threadIdx.x < 16) {
        int r = threadIdx.x;
        float mm = NEG_BIG, ss = 0.0f;
#pragma unroll
        for (int w = 0; w < WAVES_PER_BLOCK; ++w) {
            float mw = lm[w][r], sw = ls[w][r];
            float mn = fmaxf(mm, mw);
            ss = ss * expf(mm - mn) + sw * expf(mw - mn);
            mm = mn;
        }
        partM[(size_t)(row_base + r) * NCB + cb] = mm;
        partS[(size_t)(row_base + r) * NCB + cb] = ss;
    }
}

// -------- Kernel 4: combine partials + positive column, mean loss --------
__global__ void ssm_finalize_kernel(const float* __restrict__ pos_logits,
                                    const float* __restrict__ partM,
                                    const float* __restrict__ partS,
                                    float* __restrict__ out) {
    __shared__ float red[256];
    int tid = threadIdx.x;
    float acc = 0.0f;
    for (int b = tid; b < B_ROWS; b += 256) {
        float m = NEG_BIG, s = 0.0f;
#pragma unroll
        for (int k = 0; k < NCB; ++k) {
            float mk = partM[(size_t)b * NCB + k];
            float sk = partS[(size_t)b * NCB + k];
            float mn = fmaxf(m, mk);
            s = s * expf(m - mn) + sk * expf(mk - mn);
            m = mn;
        }
        float pl = pos_logits[b];
        float mn = fmaxf(m, pl);
        s = s * expf(m - mn) + expf(pl - mn);
        m = mn;
        acc += (m + logf(s)) - pl;     // logsumexp - positive logit
    }
    red[tid] = acc;
    __syncthreads();
    for (int o = 128; o > 0; o >>= 1) {
        if (tid < o) red[tid] += red[tid + o];
        __syncthreads();
    }
    if (tid == 0) out[0] = red[0] / (float)B_ROWS;
}

extern "C" void kernel_launch(void* const* d_in, const int* in_sizes, int n_in,
                              void* d_out, int out_size, void* d_ws, size_t ws_size,
                              hipStream_t stream) {
    const float* user      = (const float*)d_in[0];   // [B, D]
    const float* item_emb  = (const float*)d_in[1];   // [V, D]
    const float* item_bias = (const float*)d_in[2];   // [V]
    const int*   pos_ids   = (const int*)d_in[3];     // [B]
    const int*   samp_ids  = (const int*)d_in[4];     // [S]
    const int*   num_tries = (const int*)d_in[5];     // [1]
    float* out = (float*)d_out;

    // workspace layout
    float* semb       = (float*)d_ws;                          // S*64
    float* corr       = semb + (size_t)S_COLS * DIM;           // S
    float* pos_logits = corr + S_COLS;                         // B
    float* partM      = pos_logits + B_ROWS;                   // B*NCB
    float* partS      = partM + (size_t)B_ROWS * NCB;          // B*NCB

    ssm_gather_kernel<<<(S_COLS * 4) / 256, 256, 0, stream>>>(
        item_emb, item_bias, samp_ids, num_tries, semb, corr);

    ssm_pos_kernel<<<B_ROWS / 256, 256, 0, stream>>>(
        user, item_emb, item_bias, pos_ids, num_tries, pos_logits);

    dim3 grid(B_ROWS / 16, NCB);
    ssm_gemm_lse_kernel<<<grid, 256, 0, stream>>>(
        user, semb, corr, pos_ids, samp_ids, partM, partS);

    ssm_finalize_kernel<<<1, 256, 0, stream>>>(pos_logits, partM, partS, out);
}